// GaussianImage_Cholesky_36309653520602
// MI455X (gfx1250) — compile-verified
//
#include <hip/hip_runtime.h>
#include <hip/hip_bf16.h>

typedef __attribute__((ext_vector_type(2))) float v2f;
typedef __attribute__((ext_vector_type(8))) float v8f;
typedef __attribute__((ext_vector_type(4))) int v4i;

#define IMG_W 256
#define IMG_H 256
#define CH_G 512   // gaussians staged per LDS chunk (16 KB per buffer)

// gfx1250 async global->LDS builtins (confirmed present by previous round's
// diagnostic: param type is a b128 payload pointer, v4i*).
#if defined(__has_builtin)
#if __has_builtin(__builtin_amdgcn_global_load_async_to_lds_b128) && \
    __has_builtin(__builtin_amdgcn_s_wait_asynccnt)
#define USE_ASYNC_LDS 1
#endif
#endif
#ifndef USE_ASYNC_LDS
#define USE_ASYNC_LDS 0
#endif

#if USE_ASYNC_LDS
typedef __attribute__((address_space(1))) v4i* gv4i_ptr;
typedef __attribute__((address_space(3))) v4i* lv4i_ptr;
#endif

// ---------------------------------------------------------------------------
// Preprocess: per-gaussian params folded so the render inner loop is minimal.
// Record layout (2 x float4 = 32B per gaussian):
//   r0 = { cx, cy, 0.5*conic_a*log2e, conic_b*log2e }
//   r1 = { 0.5*conic_c*log2e, f0, f1, f2 }
// Indices [n, n_pad) are zeroed: sigma=0 -> w=1 but feats=0 -> contributes 0.
// ---------------------------------------------------------------------------
__global__ void gi_prep_kernel(const float* __restrict__ xyz,
                               const float* __restrict__ chol,
                               const float* __restrict__ fdc,
                               const float* __restrict__ rgbw,
                               float4* __restrict__ rec, int n, int n_pad) {
    int i = blockIdx.x * blockDim.x + threadIdx.x;
    if (i >= n_pad) return;
    if (i >= n) {
        rec[2 * i + 0] = make_float4(0.f, 0.f, 0.f, 0.f);
        rec[2 * i + 1] = make_float4(0.f, 0.f, 0.f, 0.f);
        return;
    }
    float tx = tanhf(xyz[2 * i + 0]);
    float ty = tanhf(xyz[2 * i + 1]);
    float cx = 0.5f * (float)IMG_W * (tx + 1.0f) - 0.5f;
    float cy = 0.5f * (float)IMG_H * (ty + 1.0f) - 0.5f;
    float l1 = chol[3 * i + 0] + 0.5f;
    float l2 = chol[3 * i + 1];
    float l3 = chol[3 * i + 2] + 0.5f;
    float s11 = l1 * l1;
    float s12 = l1 * l2;
    float s22 = l2 * l2 + l3 * l3;
    float det = s11 * s22 - s12 * s12;
    float inv = 1.0f / det;
    const float L2E = 1.4426950408889634f;  // log2(e); fold exp(-x) -> exp2(-x*L2E)
    float a2 = 0.5f * s22 * inv * L2E;
    float b2 = -s12 * inv * L2E;
    float c2 = 0.5f * s11 * inv * L2E;
    float wgt = rgbw[i];
    float f0 = fdc[3 * i + 0] * wgt;
    float f1 = fdc[3 * i + 1] * wgt;
    float f2 = fdc[3 * i + 2] * wgt;
    rec[2 * i + 0] = make_float4(cx, cy, a2, b2);
    rec[2 * i + 1] = make_float4(c2, f0, f1, f2);
}

// ---------------------------------------------------------------------------
// Render: each wave32 owns 16 consecutive pixels of one row (WMMA M dim).
// WMMA f32 16x16x4:  A[M=pixel][K=gaussian] = w,  B[K][N] cols 0..2 = RGB,
// accumulator C holds per-pixel RGB in columns 0..2.
// Lane layouts per CDNA5 ISA 7.12.2:
//   A: lane l holds M=l&15, K = 2*(l>>4) + vgpr
//   B: lane l holds N=l&15, K = 2*(l>>4) + vgpr
//   C/D: vgpr r, lane l -> row (r + 8*(l>>4)), col l&15
// Gaussian records double-buffered in LDS; next chunk streamed with
// GLOBAL_LOAD_ASYNC_TO_LDS_B128 (ASYNCcnt) while current chunk is computed.
// ---------------------------------------------------------------------------
__global__ __launch_bounds__(256) void gi_render_kernel(
    const float4* __restrict__ rec, float* __restrict__ out, int n_pad) {
    __shared__ float4 srec[2][CH_G * 2];  // 2 x 16 KB

    const int lane = threadIdx.x & 31;
    const int wv   = threadIdx.x >> 5;
    const int seg  = blockIdx.x * 8 + wv;      // 16-pixel segment id
    const int y    = seg >> 4;                 // 256/16 = 16 segments per row
    const int xb   = (seg & 15) << 4;
    const int m    = lane & 15;                // M (pixel) / N (col) index
    const int hf   = lane >> 4;                // lane-half -> K offset 2*hf

    const float pxm = (float)(xb + m);
    const float py  = (float)y;

    const int n_chunks = n_pad / CH_G;

    // stage one chunk (branch-free: rec[] is zero-padded to n_pad)
    auto stage = [&](int chunk, int buf) {
        const float4* g = rec + (size_t)chunk * (CH_G * 2);
#if USE_ASYNC_LDS
#pragma unroll
        for (int k = 0; k < (CH_G * 2) / 256; ++k) {
            int idx = (int)threadIdx.x + k * 256;
            __builtin_amdgcn_global_load_async_to_lds_b128(
                (gv4i_ptr)(g + idx), (lv4i_ptr)(&srec[buf][idx]), 0, 0);
        }
#else
#pragma unroll
        for (int k = 0; k < (CH_G * 2) / 256; ++k) {
            int idx = (int)threadIdx.x + k * 256;
            srec[buf][idx] = g[idx];
        }
#endif
    };

    v8f acc = {0.f, 0.f, 0.f, 0.f, 0.f, 0.f, 0.f, 0.f};

    stage(0, 0);
    for (int c = 0; c < n_chunks; ++c) {
        const int cur = c & 1;
#if USE_ASYNC_LDS
        __builtin_amdgcn_s_wait_asynccnt(0);
#endif
        __syncthreads();  // chunk c resident; all waves done reading buf[1-cur]
        if (c + 1 < n_chunks) stage(c + 1, 1 - cur);

        const float4* __restrict__ sb = srec[cur];

        // ---- 4 gaussians per WMMA; this wave-half supplies K = 2*hf, 2*hf+1
#pragma unroll 2
        for (int j = 0; j < CH_G; j += 4) {
            int g = j + hf * 2;
            float4 pa = sb[2 * g + 0];
            float4 qa = sb[2 * g + 1];
            float4 pb = sb[2 * g + 2];
            float4 qb = sb[2 * g + 3];

            float dxa = pxm - pa.x, dya = py - pa.y;
            float sa = dxa * (pa.z * dxa + pa.w * dya) + qa.x * dya * dya;
            float wa = __builtin_amdgcn_exp2f(-fmaxf(sa, 0.0f));

            float dxb = pxm - pb.x, dyb = py - pb.y;
            float sb2 = dxb * (pb.z * dxb + pb.w * dyb) + qb.x * dyb * dyb;
            float wb = __builtin_amdgcn_exp2f(-fmaxf(sb2, 0.0f));

            v2f A;
            A.x = wa;
            A.y = wb;

            // B: column m carries feature channel m (cols 3..15 zero)
            float ba = (m == 0) ? qa.y : (m == 1) ? qa.z : (m == 2) ? qa.w : 0.0f;
            float bb = (m == 0) ? qb.y : (m == 1) ? qb.z : (m == 2) ? qb.w : 0.0f;
            v2f B;
            B.x = ba;
            B.y = bb;

            acc = __builtin_amdgcn_wmma_f32_16x16x4_f32(
                /*neg_a=*/false, A, /*neg_b=*/false, B,
                /*c_mod=*/(short)0, acc, /*reuse_a=*/false, /*reuse_b=*/false);
        }
        // no trailing barrier needed: next iteration's stage targets buf[cur],
        // and its top-of-loop barrier orders it after this compute.
    }

    // ---- epilogue: lane col m<3 holds channel m for pixel rows r, r+8 ----
    if (m < 3) {
        const int hw = IMG_H * IMG_W;
#pragma unroll
        for (int r = 0; r < 8; ++r) {
            int row = r + 8 * hf;  // pixel offset within this 16-pixel segment
            float v = fminf(fmaxf(acc[r], 0.0f), 1.0f);
            out[m * hw + y * IMG_W + xb + row] = v;
        }
    }
}

extern "C" void kernel_launch(void* const* d_in, const int* in_sizes, int n_in,
                              void* d_out, int out_size, void* d_ws, size_t ws_size,
                              hipStream_t stream) {
    const float* xyz  = (const float*)d_in[0];  // (N,2)
    const float* chol = (const float*)d_in[1];  // (N,3)
    const float* fdc  = (const float*)d_in[2];  // (N,3)
    const float* rgbw = (const float*)d_in[3];  // (N,1)
    int n = in_sizes[0] / 2;
    int n_pad = ((n + CH_G - 1) / CH_G) * CH_G;

    float4* rec = (float4*)d_ws;  // n_pad * 32 bytes
    float* out = (float*)d_out;

    gi_prep_kernel<<<(n_pad + 255) / 256, 256, 0, stream>>>(xyz, chol, fdc, rgbw,
                                                            rec, n, n_pad);

    // 65536 pixels / (8 waves * 16 pixels) = 512 blocks
    gi_render_kernel<<<(IMG_H * IMG_W) / 128, 256, 0, stream>>>(rec, out, n_pad);
}